// SVGD_Barrier_25262997635482
// MI455X (gfx1250) — compile-verified
//
#include <hip/hip_runtime.h>
#include <math.h>

typedef __attribute__((ext_vector_type(2))) float v2f;
typedef __attribute__((ext_vector_type(8))) float v8f;

static constexpr int   NP   = 2048;          // particles
static constexpr int   DIM  = 32;            // features
static constexpr float RCON = 128.0f;        // 4*D
static constexpr int   NB   = 1024;          // histogram bins
static constexpr int   NCH  = 8;             // K-dim chunks in SVGD GEMM
static constexpr unsigned long long NN = (unsigned long long)NP * NP;

__device__ __forceinline__ float waveSum32(float v) {
#pragma unroll
  for (int s = 16; s > 0; s >>= 1) v += __shfl_xor(v, s, 32);
  return v;
}

// ---------------------------------------------------------------------------
// K0: per-particle norms, Gm = grad - x = (2*gamma/g - 2) * x ; zero maxbits
// one wave (32 lanes) per particle
// ---------------------------------------------------------------------------
__global__ void kNormsGrad(const float* __restrict__ x, const float* __restrict__ gam,
                           float* __restrict__ norms, float* __restrict__ Gm,
                           unsigned* __restrict__ maxbits) {
  if (blockIdx.x == 0 && threadIdx.x == 0) maxbits[0] = 0u;
  int gtid = blockIdx.x * blockDim.x + threadIdx.x;
  int i = gtid >> 5;
  int d = gtid & 31;
  float v  = x[i * DIM + d];
  float s  = waveSum32(v * v);          // ||x_i||^2, uniform across wave
  if (d == 0) norms[i] = s;
  float g     = s - RCON;               // < 0 in feasible region
  float gamma = gam[0];
  float coef  = 2.0f * gamma / g - 2.0f;
  Gm[i * DIM + d] = coef * v;
}

// ---------------------------------------------------------------------------
// K1: D2[i][j] = ||xi||^2 + ||xj||^2 - 2 * (xi . xj) via WMMA f32 16x16x4
// one wave per 16x16 tile; 8 WMMA steps over K=32; also atomicMax of d2
// ---------------------------------------------------------------------------
__global__ void __launch_bounds__(256) kGram(const float* __restrict__ x,
                                             const float* __restrict__ norms,
                                             float* __restrict__ D2,
                                             unsigned* __restrict__ maxbits) {
  int lane  = threadIdx.x & 31;
  int half  = lane >> 4;                 // which K pair this lane holds
  int l16   = lane & 15;
  int wid   = blockIdx.x * 8 + (threadIdx.x >> 5);   // 16384 tiles total
  int ti    = wid >> 7;                  // tile row   (128)
  int tj    = wid & 127;                 // tile col   (128)
  int rowA  = ti * 16 + l16;
  int rowB  = tj * 16 + l16;

  v8f c = {};
#pragma unroll
  for (int k0 = 0; k0 < DIM; k0 += 4) {
    int kk = k0 + half * 2;
    v2f a, b;
    a.x = x[rowA * DIM + kk];  a.y = x[rowA * DIM + kk + 1];
    b.x = x[rowB * DIM + kk];  b.y = x[rowB * DIM + kk + 1];
    c = __builtin_amdgcn_wmma_f32_16x16x4_f32(false, a, false, b,
                                              (short)0, c, false, false);
  }

  float nj = norms[tj * 16 + l16];
  float lmax = 0.0f;
#pragma unroll
  for (int r = 0; r < 8; ++r) {
    int m = r + 8 * half;                 // C layout: lanes>=16 hold M = r+8
    float d2 = norms[ti * 16 + m] + nj - 2.0f * c[r];
    d2 = fmaxf(d2, 0.0f);
    D2[(size_t)(ti * 16 + m) * NP + (tj * 16 + l16)] = d2;
    lmax = fmaxf(lmax, d2);
  }
#pragma unroll
  for (int s = 16; s > 0; s >>= 1) lmax = fmaxf(lmax, __shfl_xor(lmax, s, 32));
  if (lane == 0) atomicMax(maxbits, __float_as_uint(lmax));  // valid: d2 >= 0
}

// ---------------------------------------------------------------------------
// Median machinery: ctrl[0]=maxbits, ctrl[1]=hinv,
// sel state at ctrl[2+4*s]: lo, hi, (uint)countBelow
// ---------------------------------------------------------------------------
__global__ void kMedianInit(float* __restrict__ ctrl, unsigned* __restrict__ bins) {
  int t = threadIdx.x;
  if (t < NB) bins[t] = 0u;
  if (t == 0) {
    float mx = __uint_as_float(((unsigned*)ctrl)[0]);
    float hi = mx * 1.0000005f + 1e-6f;      // strict upper bound
    ctrl[2] = 0.0f; ctrl[3] = hi; ((unsigned*)ctrl)[4] = 0u;
    ctrl[6] = 0.0f; ctrl[7] = hi; ((unsigned*)ctrl)[8] = 0u;
  }
}

__global__ void __launch_bounds__(256) kHisto(const float* __restrict__ D2,
                                              unsigned* __restrict__ bins,
                                              const float* __restrict__ ctrl, int so) {
  __shared__ unsigned lb[NB];
  for (int t = threadIdx.x; t < NB; t += blockDim.x) lb[t] = 0u;
  __syncthreads();
  float lo = ctrl[so], hi = ctrl[so + 1];
  float scale = (float)NB / (hi - lo);
  unsigned long long stride = (unsigned long long)gridDim.x * blockDim.x;
  for (unsigned long long idx = (unsigned long long)blockIdx.x * blockDim.x + threadIdx.x;
       idx < NN; idx += stride) {
    float v = D2[idx];
    if (v >= lo && v < hi) {
      int b = (int)((v - lo) * scale);
      b = b < 0 ? 0 : (b > NB - 1 ? NB - 1 : b);
      atomicAdd(&lb[b], 1u);
    }
  }
  __syncthreads();
  for (int t = threadIdx.x; t < NB; t += blockDim.x)
    if (lb[t]) atomicAdd(&bins[t], lb[t]);
}

__global__ void kSelect(unsigned* __restrict__ bins, float* __restrict__ ctrl,
                        int so, unsigned k) {
  __shared__ unsigned sb[NB];
  int t = threadIdx.x;
  sb[t] = bins[t];
  bins[t] = 0u;                          // re-zero for next histogram pass
  __syncthreads();
  if (t == 0) {
    unsigned below = ((unsigned*)ctrl)[so + 2];
    float lo = ctrl[so], hi = ctrl[so + 1];
    float w  = (hi - lo) * (1.0f / (float)NB);
    unsigned cum = below;
    int chosen = NB - 1;
    for (int b = 0; b < NB; ++b) {
      unsigned nc = cum + sb[b];
      if (nc >= k) { chosen = b; break; }
      cum = nc;
    }
    ctrl[so]     = lo + chosen * w;
    ctrl[so + 1] = lo + (chosen + 1) * w;
    ((unsigned*)ctrl)[so + 2] = cum;
  }
}

__global__ void kFinalize(float* __restrict__ ctrl) {
  // two middle order stats of d^2 -> median of dist (sqrt is monotone)
  float mA = 0.5f * (ctrl[2] + ctrl[3]);
  float mB = 0.5f * (ctrl[6] + ctrl[7]);
  float med = 0.5f * (sqrtf(mA) + sqrtf(mB));
  float h = med / 7.6246189852f;         // log(2048)
  ctrl[1] = 1.0f / (h + 1e-6f);
}

// ---------------------------------------------------------------------------
// K3: partial K@(grad-x) + partial rowsum(K), K-dim split into NCH chunks.
// wid = ti*16 + nhalf*8 + chunk  -> 2048 waves (vs 256 unsplit): enough
// concurrency to stream the 16MB D2 matrix at HBM rate. Partials land in
// per-chunk slices; the combine is a fixed-order sum (deterministic, unlike
// float atomics).
// ---------------------------------------------------------------------------
__global__ void __launch_bounds__(256) kSvgd(const float* __restrict__ D2,
                                             const float* __restrict__ Gm,
                                             const float* __restrict__ ctrl,
                                             float* __restrict__ termP,
                                             float* __restrict__ rowsumP) {
  int lane  = threadIdx.x & 31;
  int half  = lane >> 4;
  int l16   = lane & 15;
  int wid   = (blockIdx.x * blockDim.x + threadIdx.x) >> 5;  // [0,2048)
  int ti    = wid >> 4;                  // row tile (128)
  int nhalf = (wid >> 3) & 1;            // feature half
  int chunk = wid & 7;                   // K chunk
  int n0    = nhalf * 16;
  float hinv = ctrl[1];
  int rowA = ti * 16 + l16;
  int j0 = chunk * (NP / NCH);

  v8f c = {};
  float acc = 0.0f;                      // partial row-sum of K (this chunk)
  for (int j = j0; j < j0 + NP / NCH; j += 4) {
    int jj = j + half * 2;
    float k0 = expf(-D2[(size_t)rowA * NP + jj] * hinv);
    float k1 = expf(-D2[(size_t)rowA * NP + jj + 1] * hinv);
    acc += k0 + k1;
    v2f a, b;
    a.x = k0; a.y = k1;
    b.x = Gm[jj * DIM + n0 + l16];
    b.y = Gm[(jj + 1) * DIM + n0 + l16];
    c = __builtin_amdgcn_wmma_f32_16x16x4_f32(false, a, false, b,
                                              (short)0, c, false, false);
  }
  int nIdx = n0 + l16;
#pragma unroll
  for (int r = 0; r < 8; ++r) {
    int m = r + 8 * half;
    int row = ti * 16 + m;
    termP[(size_t)chunk * (NP * DIM) + row * DIM + nIdx] = c[r];
  }
  // partial rowsum for row (ti*16 + l16): combine the two K-slice halves
  float rs = acc + __shfl_xor(acc, 16, 32);
  if (nhalf == 0 && lane < 16)
    rowsumP[chunk * NP + ti * 16 + l16] = rs;
}

// ---------------------------------------------------------------------------
// K4: deterministic combine of partials + clip + barrier freeze +
// 101-step bisection + final revert; one wave per particle
// ---------------------------------------------------------------------------
__global__ void kBarrier(const float* __restrict__ x,
                         const float* __restrict__ termP,
                         const float* __restrict__ rowsumP,
                         const float* __restrict__ gam, float* __restrict__ out) {
  int gtid = blockIdx.x * blockDim.x + threadIdx.x;
  int i = gtid >> 5;
  int d = gtid & 31;
  float xv = x[i * DIM + d];

  float term = 0.0f, rs = 0.0f;
#pragma unroll
  for (int ch = 0; ch < NCH; ++ch) {     // fixed-order sum: deterministic
    term += termP[(size_t)ch * (NP * DIM) + i * DIM + d];
    rs   += rowsumP[ch * NP + i];
  }
  float gamma = gam[0];
  float clipv = 1000.0f * gamma;
  float dx = (term + rs * xv) * (0.1f / (float)NP);
  dx = fminf(fmaxf(dx, -clipv), clipv);
  float cv = xv + dx;

  float gc = waveSum32(cv * cv) - RCON;
  bool frozen = gc > (-0.01f / gamma);
  float nv = frozen ? xv : cv;

  for (int it = 0; it < 101; ++it) {
    float gn = waveSum32(nv * nv) - RCON;
    bool viol = (gn > 0.0f) && !frozen;
    nv = viol ? 0.5f * nv + 0.5f * xv : nv;
  }
  float gn = waveSum32(nv * nv) - RCON;
  bool viol = (gn > 0.0f) && !frozen;
  out[i * DIM + d] = viol ? xv : nv;
}

// ---------------------------------------------------------------------------
extern "C" void kernel_launch(void* const* d_in, const int* in_sizes, int n_in,
                              void* d_out, int out_size, void* d_ws, size_t ws_size,
                              hipStream_t stream) {
  (void)in_sizes; (void)n_in; (void)out_size; (void)ws_size;
  const float* x   = (const float*)d_in[0];   // [2048,32]
  const float* gam = (const float*)d_in[1];   // scalar
  float* out = (float*)d_out;

  float* ws      = (float*)d_ws;
  float* norms   = ws;                                 //  2048 f
  float* Gm      = norms + NP;                         // 65536 f
  float* D2      = Gm + NP * DIM;                      // 4M f (16 MB)
  float* ctrl    = D2 + (size_t)NP * NP;               // 16 f control block
  unsigned* bins = (unsigned*)(ctrl + 16);             // 1024 u32
  float* termP   = ctrl + 16 + NB;                     // 8 * 65536 f (2 MB)
  float* rowsumP = termP + (size_t)NCH * NP * DIM;     // 8 * 2048 f

  kNormsGrad<<<256, 256, 0, stream>>>(x, gam, norms, Gm, (unsigned*)ctrl);
  kGram<<<2048, 256, 0, stream>>>(x, norms, D2, (unsigned*)ctrl);
  kMedianInit<<<1, NB, 0, stream>>>(ctrl, bins);

  const unsigned kth[2] = {2097152u, 2097153u};        // two middle order stats
  for (int sel = 0; sel < 2; ++sel) {
    int so = 2 + sel * 4;
    for (int p = 0; p < 3; ++p) {
      kHisto<<<1024, 256, 0, stream>>>(D2, bins, ctrl, so);
      kSelect<<<1, NB, 0, stream>>>(bins, ctrl, so, kth[sel]);
    }
  }
  kFinalize<<<1, 1, 0, stream>>>(ctrl);

  kSvgd<<<256, 256, 0, stream>>>(D2, Gm, ctrl, termP, rowsumP);
  kBarrier<<<256, 256, 0, stream>>>(x, termP, rowsumP, gam, out);
}